// Quantize_54288386621467
// MI455X (gfx1250) — compile-verified
//
#include <hip/hip_runtime.h>
#include <hip/hip_bf16.h>

typedef float v2f __attribute__((ext_vector_type(2)));
typedef float v8f __attribute__((ext_vector_type(8)));

#define N_SAMP 16384      // B*H*W = 16*32*32
#define DIM    128        // EMBED_DIM
#define K_EMB  10000      // N_EMBED
#define NT     625        // K_EMB / 16 column tiles

// workspace layout (byte offsets, all 16B aligned)
#define WS_CMT   0ull                       // float [K_EMB*DIM]        5,120,000 B
#define WS_BNORM 5120000ull                 // float [K_EMB]               40,000 B
#define WS_BP    5160000ull                 // float4[NT*16*32]         5,120,000 B
#define WS_IDX   10280000ull                // int   [N_SAMP]              65,536 B
#define WS_PART  10345536ull                // float [N_SAMP]              65,536 B

// ---------------------------------------------------------------------------
// Kernel 1: transpose cluster_mean [DIM,K_EMB] -> cmT [K_EMB,DIM] and compute
// per-codeword squared norms.  One block per codeword n, 128 threads = DIM.
// ---------------------------------------------------------------------------
__global__ void k_transpose_norm(const float* __restrict__ cm,
                                 float* __restrict__ cmT,
                                 float* __restrict__ bnorm) {
    int n = blockIdx.x;          // 0..K_EMB-1
    int d = threadIdx.x;         // 0..127
    float v = cm[d * K_EMB + n];
    cmT[n * DIM + d] = v;
    __shared__ float s[128];
    s[d] = v * v;
    __syncthreads();
    for (int off = 64; off >= 1; off >>= 1) {
        if (d < off) s[d] += s[d + off];
        __syncthreads();
    }
    if (d == 0) bnorm[n] = s[0];
}

// ---------------------------------------------------------------------------
// Kernel 2: repack B into WMMA 16x16x4 f32 fragment order.
// For column-tile nt, lane L holds B[K=kk..kk+1][N=nt*16+(L&15)] per k-step
// (kk = (L>>4)*2).  float4 j packs k-steps 2j and 2j+1, i.e. cmT rows
// {8j+kk, 8j+kk+1, 8j+4+kk, 8j+4+kk+1} of column n.  Stored so the GEMM loop
// reads 512B fully-coalesced b128 loads.
// ---------------------------------------------------------------------------
__global__ void k_pack_b(const float* __restrict__ cmT, float4* __restrict__ bp) {
    int t = blockIdx.x * blockDim.x + threadIdx.x;
    if (t >= NT * 16 * 32) return;
    int lane = t & 31;
    int j    = (t >> 5) & 15;
    int nt   = t >> 9;
    int n    = nt * 16 + (lane & 15);
    int kk   = (lane >> 4) * 2;
    int r0   = 8 * j + kk;
    const float* col = cmT + (size_t)n * DIM;
    float4 v;
    v.x = col[r0 + 0];
    v.y = col[r0 + 1];
    v.z = col[r0 + 4];
    v.w = col[r0 + 5];
    bp[t] = v;
}

// ---------------------------------------------------------------------------
// Kernel 3: fused fp32 WMMA GEMM + argmax.  One wave32 per 16 sample rows.
// A (16x128) is held in registers for the whole 625-tile sweep.  Per tile:
// 32 x v_wmma_f32_16x16x4_f32 on TWO independent accumulator chains
// (c0: k-steps 0..15, c1: k-steps 16..31), interleaved so back-to-back
// WMMAs never share a C/D accumulator -> no accumulate RAW stall on the
// matrix pipe.  Then fold in the codeword norm (||s||^2 is row-constant and
// drops out of argmax) and update the per-lane running maximum.  Final
// shfl_xor reduction within each 16-lane half (C layout: VGPR r = rows
// r / r+8 in low/high lane halves).
// ---------------------------------------------------------------------------
__global__ void __launch_bounds__(256)
k_gemm_argmax(const float* __restrict__ x, const float4* __restrict__ bp,
              const float* __restrict__ bnorm, int* __restrict__ idxOut,
              float* __restrict__ idxOutF) {
    int tid   = blockIdx.x * blockDim.x + threadIdx.x;
    int lane  = threadIdx.x & 31;
    int wave  = tid >> 5;              // 0..1023
    int m0    = wave * 16;
    int mlane = lane & 15;
    int kk    = (lane >> 4) * 2;

    // A fragments: lane holds A[M=m0+mlane][K=4s+kk .. +1] for s=0..31
    v2f a[32];
    const float* xrow = x + (size_t)(m0 + mlane) * DIM + kk;
#pragma unroll
    for (int s = 0; s < 32; ++s) {
        a[s].x = xrow[4 * s + 0];
        a[s].y = xrow[4 * s + 1];
    }

    float bestv[8];
    int   besti[8];
#pragma unroll
    for (int r = 0; r < 8; ++r) { bestv[r] = -__builtin_inff(); besti[r] = 0; }

    for (int nt = 0; nt < NT; ++nt) {
        const float4* bsrc = bp + (size_t)nt * 512 + lane;
        float4 b4[16];
#pragma unroll
        for (int j = 0; j < 16; ++j) b4[j] = bsrc[j * 32];
        if (nt + 1 < NT)   // locality 3 -> WGP-scope prefetch (all cache levels)
            __builtin_prefetch(bp + (size_t)(nt + 1) * 512 + lane, 0, 3);

        v8f c0 = {0.f, 0.f, 0.f, 0.f, 0.f, 0.f, 0.f, 0.f};
        v8f c1 = {0.f, 0.f, 0.f, 0.f, 0.f, 0.f, 0.f, 0.f};
#pragma unroll
        for (int s = 0; s < 16; ++s) {
            // chain 0: k-step s  (B fragment = b4[s>>1].xy/.zw)
            v2f bv0;
            if (s & 1) { bv0.x = b4[s >> 1].z; bv0.y = b4[s >> 1].w; }
            else       { bv0.x = b4[s >> 1].x; bv0.y = b4[s >> 1].y; }
            c0 = __builtin_amdgcn_wmma_f32_16x16x4_f32(
                     false, a[s], false, bv0, (short)0, c0, false, false);
            // chain 1: k-step s+16 (B fragment = b4[8+(s>>1)])
            v2f bv1;
            if (s & 1) { bv1.x = b4[8 + (s >> 1)].z; bv1.y = b4[8 + (s >> 1)].w; }
            else       { bv1.x = b4[8 + (s >> 1)].x; bv1.y = b4[8 + (s >> 1)].y; }
            c1 = __builtin_amdgcn_wmma_f32_16x16x4_f32(
                     false, a[s + 16], false, bv1, (short)0, c1, false, false);
        }

        int   n  = nt * 16 + mlane;
        float bn = bnorm[n];
#pragma unroll
        for (int r = 0; r < 8; ++r) {
            float val = fmaf(-2.0f, c0[r] + c1[r], bn);   // ||m||^2 - 2 s.m
            if (val > bestv[r]) { bestv[r] = val; besti[r] = n; }
        }
    }

    // argmax reduction across the 16 lanes of each half (n-dimension),
    // first-occurrence tie-break (smaller index wins on equality)
#pragma unroll
    for (int r = 0; r < 8; ++r) {
        float bv = bestv[r];
        int   bi = besti[r];
        for (int off = 1; off <= 8; off <<= 1) {
            float ov = __shfl_xor(bv, off, 32);
            int   oi = __shfl_xor(bi, off, 32);
            if (ov > bv || (ov == bv && oi < bi)) { bv = ov; bi = oi; }
        }
        bestv[r] = bv;
        besti[r] = bi;
    }

    if (mlane == 0) {
        int base = m0 + (lane >> 4) * 8;   // lane0 -> rows m0..m0+7, lane16 -> m0+8..m0+15
#pragma unroll
        for (int r = 0; r < 8; ++r) {
            idxOut[base + r]  = besti[r];
            idxOutF[base + r] = (float)besti[r];
        }
    }
}

// ---------------------------------------------------------------------------
// Kernel 4: gather quantize rows from cmT (coalesced) and per-row sq-diff.
// ---------------------------------------------------------------------------
__global__ void k_gather_diff(const float* __restrict__ x,
                              const float* __restrict__ cmT,
                              const int* __restrict__ idxArr,
                              float* __restrict__ qout,
                              float* __restrict__ partial) {
    int row = blockIdx.x;
    int d   = threadIdx.x;
    int idx = idxArr[row];
    float q  = cmT[(size_t)idx * DIM + d];
    float xv = x[(size_t)row * DIM + d];
    qout[(size_t)row * DIM + d] = q;
    float dd = xv - q;
    __shared__ float s[128];
    s[d] = dd * dd;
    __syncthreads();
    for (int off = 64; off >= 1; off >>= 1) {
        if (d < off) s[d] += s[d + off];
        __syncthreads();
    }
    if (d == 0) partial[row] = s[0];
}

// ---------------------------------------------------------------------------
// Kernel 5: deterministic final reduction -> mean squared diff scalar.
// ---------------------------------------------------------------------------
__global__ void k_finalize(const float* __restrict__ partial,
                           float* __restrict__ diffOut) {
    __shared__ float s[256];
    float acc = 0.f;
    for (int i = threadIdx.x; i < N_SAMP; i += 256) acc += partial[i];
    s[threadIdx.x] = acc;
    __syncthreads();
    for (int off = 128; off >= 1; off >>= 1) {
        if (threadIdx.x < off) s[threadIdx.x] += s[threadIdx.x + off];
        __syncthreads();
    }
    if (threadIdx.x == 0)
        diffOut[0] = s[0] * (1.0f / (float)(N_SAMP * DIM));
}

extern "C" void kernel_launch(void* const* d_in, const int* in_sizes, int n_in,
                              void* d_out, int out_size, void* d_ws, size_t ws_size,
                              hipStream_t stream) {
    (void)in_sizes; (void)n_in; (void)out_size; (void)ws_size;

    const float* x  = (const float*)d_in[0];   // [16,32,32,128]
    const float* cm = (const float*)d_in[1];   // [128,10000]
    float* out = (float*)d_out;
    char*  ws  = (char*)d_ws;

    float*  cmT     = (float*)(ws + WS_CMT);
    float*  bnorm   = (float*)(ws + WS_BNORM);
    float4* bp      = (float4*)(ws + WS_BP);
    int*    idxArr  = (int*)(ws + WS_IDX);
    float*  partial = (float*)(ws + WS_PART);

    float* qout    = out;                         // [N_SAMP*DIM]
    float* idxF    = out + (size_t)N_SAMP * DIM;  // [N_SAMP] (indices as float)
    float* diffOut = idxF + N_SAMP;               // [1]

    k_transpose_norm<<<K_EMB, 128, 0, stream>>>(cm, cmT, bnorm);
    k_pack_b<<<(NT * 16 * 32 + 255) / 256, 256, 0, stream>>>(cmT, bp);
    k_gemm_argmax<<<128, 256, 0, stream>>>(x, bp, bnorm, idxArr, idxF);
    k_gather_diff<<<N_SAMP, 128, 0, stream>>>(x, cmT, idxArr, qout, partial);
    k_finalize<<<1, 256, 0, stream>>>(partial, diffOut);
}